// AbsoluteMinimalBlock_12068858102142
// MI455X (gfx1250) — compile-verified
//
#include <hip/hip_runtime.h>
#include <hip/hip_bf16.h>
#include <math.h>

// Problem constants (match reference): B=4, S=8192, D=1024, R_SSM=1, R_FFN=2
#define BATCH 4
#define SEQ   8192
#define DIM   1024
#define ROWS  (BATCH * SEQ)   // 32768 tokens
#define EPSF  1e-6f

typedef __attribute__((ext_vector_type(2))) float v2f;
typedef __attribute__((ext_vector_type(8))) float v8f;

__device__ __forceinline__ float gelu_tanh(float v) {
    // jax.nn.gelu default (approximate=True, tanh form)
    const float k0 = 0.7978845608028654f;   // sqrt(2/pi)
    const float k1 = 0.044715f;
    float t = tanhf(k0 * (v + k1 * v * v * v));
    return 0.5f * v * (1.0f + t);
}

// -----------------------------------------------------------------------------
// Kernel 1: per-token RMS stats + rank-1 input projection via V_WMMA_F32_16X16X4_F32
// One wave (32 lanes) owns a 16-row tile.
//   A layout (16x4 f32): lanes 0-15 => M=lane, K={k,k+1}; lanes 16-31 => M=lane-16, K={k+2,k+3}
//   B layout (4x16 f32): lane n = column n; column 0 carries wv = norm_w * W_in
//   C/D (16x16 f32): VGPR r: lanes 0-15 -> M=r, lanes 16-31 -> M=r+8; N = lane&15
// B is built with an UNCONDITIONAL LDS read masked by a per-lane selector so the
// inner loop has no EXEC save/restore and can be software-pipelined.
// dotw[row] = sum_d x[row,d] * wv[d]   (rsqrt scaling applied in the scan kernel)
// msout[row] = sum_d x[row,d]^2
// -----------------------------------------------------------------------------
__global__ __launch_bounds__(256)
void uprep_kernel(const float* __restrict__ x, const float* __restrict__ norm_w,
                  const float* __restrict__ w_in,
                  float* __restrict__ dotw, float* __restrict__ msout) {
    __shared__ float s_wv[DIM];
    for (int i = threadIdx.x; i < DIM; i += 256)
        s_wv[i] = norm_w[i] * w_in[i];           // fold norm weight into W_in
    __syncthreads();

    const int lane = threadIdx.x & 31;
    const int wave = threadIdx.x >> 5;
    const int m    = lane & 15;                   // row within tile
    const int hi   = lane >> 4;                   // K-half select
    const int row0 = (blockIdx.x * 8 + wave) * 16;
    const float* xrow = x + (size_t)(row0 + m) * DIM;
    const float bsel = (m == 0) ? 1.0f : 0.0f;    // only column 0 of B is live

    v8f c = {};
    float ssa = 0.0f, ssb = 0.0f;                 // two accumulators: break FMA chain
    #pragma unroll 8
    for (int k = 0; k < DIM; k += 4) {
        const int kk = k + hi * 2;
        float2 xv = *(const float2*)(xrow + kk);
        v2f a; a.x = xv.x; a.y = xv.y;
        v2f b; b.x = s_wv[kk] * bsel; b.y = s_wv[kk + 1] * bsel;
        c = __builtin_amdgcn_wmma_f32_16x16x4_f32(false, a, false, b,
                                                  (short)0, c, false, false);
        ssa = fmaf(xv.x, xv.x, ssa);
        ssb = fmaf(xv.y, xv.y, ssb);
    }
    float ss = ssa + ssb;
    // combine the two K-halves of each row's sum of squares
    ss += __shfl_xor(ss, 16, 32);
    if (hi == 0) msout[row0 + m] = ss;
    // column N=0 of D lives in lanes 0 (M=0..7) and 16 (M=8..15)
    if (m == 0) {
        #pragma unroll
        for (int r = 0; r < 8; ++r) dotw[row0 + hi * 8 + r] = c[r];
    }
}

// -----------------------------------------------------------------------------
// Kernel 2: first-order linear recurrence h_t = a*h_{t-1} + u_t with constant a.
// Hillis-Steele scan in LDS: h_i += a^(2^s) * h_{i-2^s}. One block per batch,
// 8 chunks of 1024. Removes the 8192-long serial dependency.
// u_t = dotw_t * rsqrt(ms_t/D + eps)
// -----------------------------------------------------------------------------
__global__ __launch_bounds__(1024)
void scan_kernel(const float* __restrict__ dotw, const float* __restrict__ ms,
                 const float* __restrict__ a_log, float* __restrict__ hs) {
    __shared__ float sh[1024];
    const int b = blockIdx.x;
    const int t = threadIdx.x;
    const float a = 1.0f / (1.0f + __expf(-a_log[0]));   // sigmoid, in (0,1)
    const float apow_t1 = __expf(__logf(a) * (float)(t + 1)); // a^(t+1) for carry
    float carry = 0.0f;

    for (int chunk = 0; chunk < SEQ; chunk += 1024) {
        const size_t idx = (size_t)b * SEQ + chunk + t;
        float u = dotw[idx] * rsqrtf(ms[idx] * (1.0f / DIM) + EPSF);
        sh[t] = u;
        __syncthreads();
        float h  = u;
        float ao = a;                                    // a^(offset)
        for (int off = 1; off < 1024; off <<= 1) {
            float left = (t >= off) ? sh[t - off] : 0.0f;
            __syncthreads();
            h = fmaf(ao, left, h);
            sh[t] = h;
            ao = ao * ao;
            __syncthreads();
        }
        h = fmaf(apow_t1, carry, h);                     // fold in previous chunk
        hs[idx] = h;
        if (t == 1023) sh[0] = h;
        __syncthreads();
        carry = sh[0];
        __syncthreads();
    }
}

// -----------------------------------------------------------------------------
// Kernel 3: fused SSM output + residual + RMSNorm2 + rank-2 FFN + residual.
//   newx = x + h*W_out;  ms2 = sum newx^2
//   raw ffn dots via WMMA: B columns 0/1 = norm_w * ffn_w1[:,r]
//   g_r = gelu(dot_r * rsqrt(ms2/D+eps));  out = newx + g0*w2[0] + g1*w2[1]
// Second loop recomputes newx (x re-read is L2-warm; 192MB L2).
// B operand: unconditional LDS read from a hoisted per-lane column pointer,
// masked by value -> no divergent loads in the WMMA loop.
// -----------------------------------------------------------------------------
__global__ __launch_bounds__(256)
void fused_out_kernel(const float* __restrict__ x, const float* __restrict__ hs,
                      const float* __restrict__ norm_w, const float* __restrict__ w_out,
                      const float* __restrict__ f1, const float* __restrict__ f2,
                      float* __restrict__ out) {
    __shared__ float s_wo[DIM];
    __shared__ float s_wf1[2][DIM];
    __shared__ float s_w2[2][DIM];
    __shared__ float s_ms[8][16];
    __shared__ float s_d[8][2][16];

    for (int i = threadIdx.x; i < DIM; i += 256) {
        s_wo[i] = w_out[i];
        const float nw = norm_w[i];
        s_wf1[0][i] = nw * f1[i * 2 + 0];   // ffn_w1 is [D,2] row-major
        s_wf1[1][i] = nw * f1[i * 2 + 1];
        s_w2[0][i]  = f2[i];                // ffn_w2 is [2,D]
        s_w2[1][i]  = f2[DIM + i];
    }
    __syncthreads();

    const int lane = threadIdx.x & 31;
    const int wave = threadIdx.x >> 5;
    const int m    = lane & 15;
    const int hi   = lane >> 4;
    const int row0 = (blockIdx.x * 8 + wave) * 16;
    const float* xrow = x + (size_t)(row0 + m) * DIM;
    const float hrow  = hs[row0 + m];
    const float bsel  = (m < 2) ? 1.0f : 0.0f;          // columns 0,1 live
    const float* bptr = s_wf1[(m < 2) ? m : 0];         // hoisted column base

    v8f c = {};
    float ssa = 0.0f, ssb = 0.0f;
    #pragma unroll 8
    for (int k = 0; k < DIM; k += 4) {
        const int kk = k + hi * 2;
        float2 xv = *(const float2*)(xrow + kk);
        float nx0 = fmaf(hrow, s_wo[kk],     xv.x);
        float nx1 = fmaf(hrow, s_wo[kk + 1], xv.y);
        v2f a; a.x = nx0; a.y = nx1;
        v2f b; b.x = bptr[kk] * bsel; b.y = bptr[kk + 1] * bsel;
        c = __builtin_amdgcn_wmma_f32_16x16x4_f32(false, a, false, b,
                                                  (short)0, c, false, false);
        ssa = fmaf(nx0, nx0, ssa);
        ssb = fmaf(nx1, nx1, ssb);
    }
    float ss2 = ssa + ssb;
    ss2 += __shfl_xor(ss2, 16, 32);
    if (hi == 0) s_ms[wave][m] = ss2;
    if (m < 2) {                 // lanes N=0,1 hold ffn dot columns
        #pragma unroll
        for (int r = 0; r < 8; ++r) s_d[wave][m][hi * 8 + r] = c[r];
    }
    __syncthreads();

    const float inv2 = rsqrtf(s_ms[wave][m] * (1.0f / DIM) + EPSF);
    const float g0 = gelu_tanh(s_d[wave][0][m] * inv2);
    const float g1 = gelu_tanh(s_d[wave][1][m] * inv2);

    float* orow = out + (size_t)(row0 + m) * DIM;
    #pragma unroll 4
    for (int k = 0; k < DIM; k += 4) {
        const int kk = k + hi * 2;
        float2 xv = *(const float2*)(xrow + kk);
        float nx0 = fmaf(hrow, s_wo[kk],     xv.x);
        float nx1 = fmaf(hrow, s_wo[kk + 1], xv.y);
        float2 ov;
        ov.x = nx0 + g0 * s_w2[0][kk]     + g1 * s_w2[1][kk];
        ov.y = nx1 + g0 * s_w2[0][kk + 1] + g1 * s_w2[1][kk + 1];
        *(float2*)(orow + kk) = ov;
    }
}

// -----------------------------------------------------------------------------
// Launch: inputs in setup_inputs() order:
//   0:x [B,S,D] 1:norm_w [D] 2:W_in [D,1] 3:a_log [1] 4:W_out [1,D]
//   5:ffn_w1 [D,2] 6:ffn_w2 [2,D]   -> out fp32 [B,S,D]
// Workspace: dotw[ROWS] | ms[ROWS] | hs[ROWS]  (384 KB)
// -----------------------------------------------------------------------------
extern "C" void kernel_launch(void* const* d_in, const int* in_sizes, int n_in,
                              void* d_out, int out_size, void* d_ws, size_t ws_size,
                              hipStream_t stream) {
    const float* x      = (const float*)d_in[0];
    const float* norm_w = (const float*)d_in[1];
    const float* w_in   = (const float*)d_in[2];
    const float* a_log  = (const float*)d_in[3];
    const float* w_out  = (const float*)d_in[4];
    const float* f1     = (const float*)d_in[5];
    const float* f2     = (const float*)d_in[6];
    float* out = (float*)d_out;

    float* ws   = (float*)d_ws;
    float* dotw = ws;
    float* ms   = ws + ROWS;
    float* hs   = ws + 2 * ROWS;

    // 2048 sixteen-row tiles, 8 waves (tiles) per 256-thread block -> 256 blocks
    uprep_kernel<<<ROWS / 16 / 8, 256, 0, stream>>>(x, norm_w, w_in, dotw, ms);
    scan_kernel<<<BATCH, 1024, 0, stream>>>(dotw, ms, a_log, hs);
    fused_out_kernel<<<ROWS / 16 / 8, 256, 0, stream>>>(x, hs, norm_w, w_out, f1, f2, out);
}